// DepthNet_13778255085984
// MI455X (gfx1250) — compile-verified
//
#include <hip/hip_runtime.h>

typedef float v2f __attribute__((ext_vector_type(2)));
typedef float v8f __attribute__((ext_vector_type(8)));

namespace {
constexpr int kB = 2, kV = 5, kC = 32, kD = 48, kH = 128, kW = 160;
constexpr int kHW = kH * kW;
// cost-volume tile
constexpr int TD = 4, TH = 8, TW = 16;
constexpr int HTD = TD + 2, HTH = TH + 2, HTW = TW + 2;
constexpr int NHALO = HTD * HTH * HTW;   // 1080 halo voxels
constexpr int VS = 33;                   // padded channel stride (bank spread)
constexpr int WS_COST_OFF = 128;         // floats: [0..95] transforms, then cost
}

// ---------------- kernel 1: projection transforms ----------------
__device__ static void inv4(const float m[16], float o[16]) {
  float inv[16];
  inv[0]  =  m[5]*m[10]*m[15] - m[5]*m[11]*m[14] - m[9]*m[6]*m[15] + m[9]*m[7]*m[14] + m[13]*m[6]*m[11] - m[13]*m[7]*m[10];
  inv[4]  = -m[4]*m[10]*m[15] + m[4]*m[11]*m[14] + m[8]*m[6]*m[15] - m[8]*m[7]*m[14] - m[12]*m[6]*m[11] + m[12]*m[7]*m[10];
  inv[8]  =  m[4]*m[9]*m[15]  - m[4]*m[11]*m[13] - m[8]*m[5]*m[15] + m[8]*m[7]*m[13] + m[12]*m[5]*m[11] - m[12]*m[7]*m[9];
  inv[12] = -m[4]*m[9]*m[14]  + m[4]*m[10]*m[13] + m[8]*m[5]*m[14] - m[8]*m[6]*m[13] - m[12]*m[5]*m[10] + m[12]*m[6]*m[9];
  inv[1]  = -m[1]*m[10]*m[15] + m[1]*m[11]*m[14] + m[9]*m[2]*m[15] - m[9]*m[3]*m[14] - m[13]*m[2]*m[11] + m[13]*m[3]*m[10];
  inv[5]  =  m[0]*m[10]*m[15] - m[0]*m[11]*m[14] - m[8]*m[2]*m[15] + m[8]*m[3]*m[14] + m[12]*m[2]*m[11] - m[12]*m[3]*m[10];
  inv[9]  = -m[0]*m[9]*m[15]  + m[0]*m[11]*m[13] + m[8]*m[1]*m[15] - m[8]*m[3]*m[13] - m[12]*m[1]*m[11] + m[12]*m[3]*m[9];
  inv[13] =  m[0]*m[9]*m[14]  - m[0]*m[10]*m[13] - m[8]*m[1]*m[14] + m[8]*m[2]*m[13] + m[12]*m[1]*m[10] - m[12]*m[2]*m[9];
  inv[2]  =  m[1]*m[6]*m[15]  - m[1]*m[7]*m[14]  - m[5]*m[2]*m[15] + m[5]*m[3]*m[14] + m[13]*m[2]*m[7]  - m[13]*m[3]*m[6];
  inv[6]  = -m[0]*m[6]*m[15]  + m[0]*m[7]*m[14]  + m[4]*m[2]*m[15] - m[4]*m[3]*m[14] - m[12]*m[2]*m[7]  + m[12]*m[3]*m[6];
  inv[10] =  m[0]*m[5]*m[15]  - m[0]*m[7]*m[13]  - m[4]*m[1]*m[15] + m[4]*m[3]*m[13] + m[12]*m[1]*m[7]  - m[12]*m[3]*m[5];
  inv[14] = -m[0]*m[5]*m[14]  + m[0]*m[6]*m[13]  + m[4]*m[1]*m[14] - m[4]*m[2]*m[13] - m[12]*m[1]*m[6]  + m[12]*m[2]*m[5];
  inv[3]  = -m[1]*m[6]*m[11]  + m[1]*m[7]*m[10]  + m[5]*m[2]*m[11] - m[5]*m[3]*m[10] - m[9]*m[2]*m[7]   + m[9]*m[3]*m[6];
  inv[7]  =  m[0]*m[6]*m[11]  - m[0]*m[7]*m[10]  - m[4]*m[2]*m[11] + m[4]*m[3]*m[10] + m[8]*m[2]*m[7]   - m[8]*m[3]*m[6];
  inv[11] = -m[0]*m[5]*m[11]  + m[0]*m[7]*m[9]   + m[4]*m[1]*m[11] - m[4]*m[3]*m[9]  - m[8]*m[1]*m[7]   + m[8]*m[3]*m[5];
  inv[15] =  m[0]*m[5]*m[10]  - m[0]*m[6]*m[9]   - m[4]*m[1]*m[10] + m[4]*m[2]*m[9]  + m[8]*m[1]*m[6]   - m[8]*m[2]*m[5];
  float det = m[0]*inv[0] + m[1]*inv[4] + m[2]*inv[8] + m[3]*inv[12];
  det = 1.0f / det;
  #pragma unroll
  for (int i = 0; i < 16; i++) o[i] = inv[i] * det;
}

__global__ void k_transforms(const float* __restrict__ proj, float* __restrict__ ws) {
  int b = threadIdx.x;
  if (b >= kB) return;
  float ref[16], invr[16];
  // combine(E, K): top = K[:3,:3] @ E[:3,:4]; bottom = E[3,:]
  auto combine = [&](int v, float* out) {
    const float* Ep = proj + (((b * kV + v) * 2 + 0) * 16);
    const float* Kp = proj + (((b * kV + v) * 2 + 1) * 16);
    #pragma unroll
    for (int i = 0; i < 3; i++)
      #pragma unroll
      for (int j = 0; j < 4; j++) {
        float a = 0.f;
        #pragma unroll
        for (int k = 0; k < 3; k++) a += Kp[i*4+k] * Ep[k*4+j];
        out[i*4+j] = a;
      }
    #pragma unroll
    for (int j = 0; j < 4; j++) out[12+j] = Ep[12+j];
  };
  combine(0, ref);
  inv4(ref, invr);
  for (int v = 1; v < kV; v++) {
    float src[16], T[16];
    combine(v, src);
    #pragma unroll
    for (int i = 0; i < 4; i++)
      #pragma unroll
      for (int j = 0; j < 4; j++) {
        float a = 0.f;
        #pragma unroll
        for (int k = 0; k < 4; k++) a += src[i*4+k] * invr[k*4+j];
        T[i*4+j] = a;
      }
    float* o = ws + (b * 4 + (v - 1)) * 12;
    #pragma unroll
    for (int i = 0; i < 3; i++)
      #pragma unroll
      for (int j = 0; j < 3; j++) o[i*3+j] = T[i*4+j];
    o[9] = T[3]; o[10] = T[7]; o[11] = T[11];
  }
}

// ------- kernel 2: fused warp + variance (LDS) + 3x3x3 conv via WMMA -------
__global__ __launch_bounds__(256) void k_costvol(
    const float* __restrict__ feat, const float* __restrict__ dv,
    const float* __restrict__ wreg, const float* __restrict__ wsT,
    float* __restrict__ cost) {
  extern __shared__ float smem[];
  float* lds_w   = smem;          // 864: weights reordered [tap*32 + c]
  float* lds_T   = smem + 864;    // 48:  4 views x (rot9 + trans3)
  float* lds_var = smem + 912;    // NHALO * VS

  const int tid = threadIdx.x;
  const int wt = blockIdx.x, ht = blockIdx.y;
  const int b  = blockIdx.z / (kD / TD);
  const int dt = blockIdx.z % (kD / TD);
  const int d0 = dt * TD, h0 = ht * TH, w0 = wt * TW;

  for (int i = tid; i < 864; i += 256) {
    int t = i >> 5, c = i & 31;
    lds_w[i] = wreg[c * 27 + t];          // w_reg (1,C,3,3,3): c*27 + tap
  }
  if (tid < 48) lds_T[tid] = wsT[b * 48 + tid];
  __syncthreads();

  // -------- Phase A: variance volume for halo tile into LDS --------
  const float* fb = feat + b * (kV * kC * kHW);
  for (int idx = tid; idx < NHALO; idx += 256) {
    const int lz  = idx / (HTH * HTW);
    const int rem = idx % (HTH * HTW);
    const int ly  = rem / HTW;
    const int lx  = rem % HTW;
    const int dd = d0 + lz - 1, hh = h0 + ly - 1, ww = w0 + lx - 1;
    float* vout = lds_var + idx * VS;
    if (dd < 0 || dd >= kD || hh < 0 || hh >= kH || ww < 0 || ww >= kW) {
      #pragma unroll 8
      for (int c = 0; c < kC; c++) vout[c] = 0.f;
      continue;
    }
    const float depth = dv[b * kD + dd];
    int   cidx[4][4];
    float cw[4][4];
    #pragma unroll
    for (int v = 0; v < 4; v++) {
      const float* T = lds_T + v * 12;
      const float fx = T[0]*ww + T[1]*hh + T[2];
      const float fy = T[3]*ww + T[4]*hh + T[5];
      const float fz = T[6]*ww + T[7]*hh + T[8];
      const float pz = fz * depth + T[11];
      const float rz = 1.f / pz;
      const float ix = (fx * depth + T[9])  * rz;
      const float iy = (fy * depth + T[10]) * rz;
      const float x0 = floorf(ix), y0 = floorf(iy);
      const float wx1 = ix - x0, wy1 = iy - y0;
      const float wx0 = 1.f - wx1, wy0 = 1.f - wy1;
      const float x1 = x0 + 1.f, y1 = y0 + 1.f;
      auto mk = [&](float xf, float yf, int slot, float wgt) {
        bool valid = (xf >= 0.f) && (xf <= (float)(kW - 1)) &&
                     (yf >= 0.f) && (yf <= (float)(kH - 1));
        int xi = (int)fminf(fmaxf(xf, 0.f), (float)(kW - 1));
        int yi = (int)fminf(fmaxf(yf, 0.f), (float)(kH - 1));
        cidx[v][slot] = yi * kW + xi;
        cw[v][slot]   = valid ? wgt : 0.f;
      };
      mk(x0, y0, 0, wx0 * wy0);
      mk(x1, y0, 1, wx1 * wy0);
      mk(x0, y1, 2, wx0 * wy1);
      mk(x1, y1, 3, wx1 * wy1);
    }
    const int pix = hh * kW + ww;
    #pragma unroll 4
    for (int c = 0; c < kC; c++) {
      const float r = fb[c * kHW + pix];           // reference view 0
      float s = r, q = r * r;
      #pragma unroll
      for (int v = 0; v < 4; v++) {
        const float* fv = fb + (v + 1) * (kC * kHW) + c * kHW;
        float smp = fv[cidx[v][0]] * cw[v][0] + fv[cidx[v][1]] * cw[v][1]
                  + fv[cidx[v][2]] * cw[v][2] + fv[cidx[v][3]] * cw[v][3];
        s += smp;
        q += smp * smp;
      }
      const float sm = s * (1.f / kV);
      vout[c] = q * (1.f / kV) - sm * sm;
    }
  }
  __syncthreads();

  // -------- Phase B: 3x3x3xC conv via V_WMMA_F32_16X16X4_F32 --------
  // M = 16 pixels (one W-row of the tile), K = (tap, channel) in steps of 4,
  // B = weight vector replicated across all 16 N columns.
  const int lane = tid & 31;
  const int wid  = tid >> 5;
  const int m    = lane & 15;          // A-matrix row = pixel index
  const int hi   = lane >> 4;          // half-wave selects K-pair {0,1} vs {2,3}
  const int hi2  = hi * 2;

  for (int g = wid; g < TD * TH; g += 8) {
    const int dz = g / TH, hy = g % TH;
    v8f acc = {0.f, 0.f, 0.f, 0.f, 0.f, 0.f, 0.f, 0.f};
    #pragma unroll 1
    for (int t = 0; t < 27; t++) {
      const int kd = t / 9, kh = (t % 9) / 3, kw = t % 3;
      const int vox = ((dz + kd) * HTH + (hy + kh)) * HTW + (m + kw);
      const float* ap = lds_var + vox * VS + hi2;
      const float* bp = lds_w + t * 32 + hi2;
      #pragma unroll
      for (int cb = 0; cb < 32; cb += 4) {
        v2f a, bv;
        a.x  = ap[cb];     a.y  = ap[cb + 1];
        bv.x = bp[cb];     bv.y = bp[cb + 1];
        acc = __builtin_amdgcn_wmma_f32_16x16x4_f32(
            false, a, false, bv, (short)0, acc, false, false);
      }
    }
    // D f32 layout: lane 0 holds M=0..7 (vgpr r -> M=r), lane 16 holds M=8..15.
    if (m == 0) {
      float* cp = cost + ((b * kD + d0 + dz) * kH + (h0 + hy)) * kW + w0 + hi * 8;
      #pragma unroll
      for (int r = 0; r < 8; r++) cp[r] = acc[r];
    }
  }
}

// ------- kernel 3: softmax over D, depth / conf / itg outputs -------
__global__ __launch_bounds__(256) void k_post(const float* __restrict__ cost,
                                              const float* __restrict__ dv,
                                              float* __restrict__ out) {
  const int i = blockIdx.x * blockDim.x + threadIdx.x;
  if (i >= kB * kHW) return;
  const int b = i / kHW, pix = i % kHW;
  float p[kD];
  float mx = -3.402823e38f;
  #pragma unroll
  for (int d = 0; d < kD; d++) {
    p[d] = cost[(b * kD + d) * kHW + pix];
    mx = fmaxf(mx, p[d]);
  }
  float s = 0.f;
  #pragma unroll
  for (int d = 0; d < kD; d++) { p[d] = __expf(p[d] - mx); s += p[d]; }
  const float invs = 1.f / s;
  float sabs = 0.f;
  #pragma unroll
  for (int d = 0; d < kD; d++) { p[d] *= invs; sabs += fabsf(p[d]); }
  const float itgs = 1.f / fmaxf(sabs, 1e-12f);
  float depth = 0.f, dif = 0.f;
  #pragma unroll
  for (int d = 0; d < kD; d++) {
    const float it = p[d] * itgs;
    depth += it * dv[b * kD + d];
    dif   += p[d] * (float)d;
    out[2 * kB * kHW + (b * kD + d) * kHW + pix] = it;   // itg
  }
  int di = (int)dif;                       // truncation, matches astype(int32)
  di = di < 0 ? 0 : (di > kD - 1 ? kD - 1 : di);
  float conf = 0.f;
  #pragma unroll
  for (int d = 0; d < kD; d++) {
    const bool inwin = (d >= di - 1) && (d <= di + 2);
    conf += inwin ? p[d] : 0.f;
  }
  out[i] = depth;                          // depth
  out[kB * kHW + i] = conf;                // conf
}

extern "C" void kernel_launch(void* const* d_in, const int* in_sizes, int n_in,
                              void* d_out, int out_size, void* d_ws, size_t ws_size,
                              hipStream_t stream) {
  (void)in_sizes; (void)n_in; (void)out_size; (void)ws_size;
  const float* feat = (const float*)d_in[0];   // (B,V,C,H,W)
  const float* proj = (const float*)d_in[1];   // (B,V,2,4,4)
  const float* dv   = (const float*)d_in[2];   // (B,D)
  const float* wreg = (const float*)d_in[3];   // (1,C,3,3,3)
  float* out = (float*)d_out;
  float* ws  = (float*)d_ws;
  float* cost = ws + WS_COST_OFF;              // B*D*H*W floats (~7.9 MB)

  k_transforms<<<1, 64, 0, stream>>>(proj, ws);

  dim3 grid2(kW / TW, kH / TH, kB * (kD / TD));         // (10, 16, 24)
  size_t smem = (size_t)(864 + 48 + NHALO * VS) * sizeof(float);  // ~143 KB
  k_costvol<<<grid2, 256, smem, stream>>>(feat, dv, wreg, ws, cost);

  const int n3 = kB * kHW;
  k_post<<<(n3 + 255) / 256, 256, 0, stream>>>(cost, dv, out);
}